// RecurrentGCN_26164940767928
// MI455X (gfx1250) — compile-verified
//
#include <hip/hip_runtime.h>

#define HC 4      // H_CONV / F_IN
#define HL 32     // LSTM hidden

typedef __attribute__((ext_vector_type(2))) float v2f;
typedef __attribute__((ext_vector_type(8))) float v8f;

// gfx1250 has native V_TANH_F32 (single trans-op). Use it when the toolchain
// declares the builtin; otherwise a branch-free exp-based fallback.
#if __has_builtin(__builtin_amdgcn_tanhf)
__device__ __forceinline__ float fast_tanh(float x) {
    return __builtin_amdgcn_tanhf(x);
}
#else
__device__ __forceinline__ float fast_tanh(float x) {
    float e = __expf(2.0f * x);
    return 1.0f - 2.0f / (e + 1.0f);
}
#endif

// Exact identity: sigmoid(x) = 0.5*(1 + tanh(x/2))
__device__ __forceinline__ float fast_sigmoid(float x) {
    return 0.5f * fast_tanh(0.5f * x) + 0.5f;
}

// ---------------------------------------------------------------------------
// Kernel 1: m = x @ ggc_w  (N x 4) @ (4 x 4), and zero agg/cnt accumulators.
// ---------------------------------------------------------------------------
__global__ void k_ggc_init(const float* __restrict__ x,
                           const float* __restrict__ ggc_w,
                           float* __restrict__ m,
                           float* __restrict__ agg,
                           float* __restrict__ cnt, int n) {
    int i = blockIdx.x * blockDim.x + threadIdx.x;
    if (i >= n) return;
    float4 xi = ((const float4*)x)[i];
    float w[16];
#pragma unroll
    for (int t = 0; t < 16; ++t) w[t] = ggc_w[t];   // row-major [4][4]
    float4 mo;
    mo.x = xi.x * w[0] + xi.y * w[4] + xi.z * w[8]  + xi.w * w[12];
    mo.y = xi.x * w[1] + xi.y * w[5] + xi.z * w[9]  + xi.w * w[13];
    mo.z = xi.x * w[2] + xi.y * w[6] + xi.z * w[10] + xi.w * w[14];
    mo.w = xi.x * w[3] + xi.y * w[7] + xi.z * w[11] + xi.w * w[15];
    ((float4*)m)[i]   = mo;
    ((float4*)agg)[i] = make_float4(0.f, 0.f, 0.f, 0.f);
    cnt[i] = 0.f;
}

// ---------------------------------------------------------------------------
// Kernel 2: edge gather + weighted scatter-add (mean numerator) + degree cnt.
// Memory-bound phase: ~6.4M edges; atomics resolve in L2 (agg/cnt are 2 MB).
// ---------------------------------------------------------------------------
__global__ void k_edge_scatter(const long long* __restrict__ src,
                               const long long* __restrict__ dst,
                               const float* __restrict__ ew,
                               const float* __restrict__ m,
                               float* __restrict__ agg,
                               float* __restrict__ cnt, int e) {
    int i = blockIdx.x * blockDim.x + threadIdx.x;
    if (i >= e) return;
    int s = (int)src[i];
    int d = (int)dst[i];
    float w = ew[i];
    float4 ms = ((const float4*)m)[s];
    atomicAdd(&agg[d * 4 + 0], ms.x * w);
    atomicAdd(&agg[d * 4 + 1], ms.y * w);
    atomicAdd(&agg[d * 4 + 2], ms.z * w);
    atomicAdd(&agg[d * 4 + 3], ms.w * w);
    atomicAdd(&cnt[d], 1.0f);
}

// ---------------------------------------------------------------------------
// Kernel 3: mean-normalize + GRUCell(agg, x) -> h_tilde  (all 4-wide, scalar)
// ---------------------------------------------------------------------------
__global__ void k_gru(const float* __restrict__ x,
                      const float* __restrict__ agg,
                      const float* __restrict__ cnt,
                      const float* __restrict__ w_ih,   // [12,4]
                      const float* __restrict__ w_hh,   // [12,4]
                      const float* __restrict__ b_ih,   // [12]
                      const float* __restrict__ b_hh,   // [12]
                      float* __restrict__ ht, int n) {
    int i = blockIdx.x * blockDim.x + threadIdx.x;
    if (i >= n) return;
    float inv = 1.0f / fmaxf(cnt[i], 1.0f);
    float4 a4 = ((const float4*)agg)[i];
    float a[4] = {a4.x * inv, a4.y * inv, a4.z * inv, a4.w * inv};
    float4 x4 = ((const float4*)x)[i];
    float xv[4] = {x4.x, x4.y, x4.z, x4.w};

    float gi[12], gh[12];
#pragma unroll
    for (int j = 0; j < 12; ++j) {
        float si = b_ih[j], sh = b_hh[j];
#pragma unroll
        for (int k = 0; k < 4; ++k) {
            si += a[k]  * w_ih[j * 4 + k];
            sh += xv[k] * w_hh[j * 4 + k];
        }
        gi[j] = si; gh[j] = sh;
    }
    float o[4];
#pragma unroll
    for (int k = 0; k < 4; ++k) {
        float r  = fast_sigmoid(gi[k]     + gh[k]);
        float z  = fast_sigmoid(gi[4 + k] + gh[4 + k]);
        float nc = fast_tanh(gi[8 + k] + r * gh[8 + k]);
        o[k] = (1.0f - z) * nc + z * xv[k];
    }
    ((float4*)ht)[i] = make_float4(o[0], o[1], o[2], o[3]);
}

// ---------------------------------------------------------------------------
// Kernel 4: LSTM step (h0=c0=0) + relu + Linear(32,1) via V_WMMA_F32_16X16X4_F32
// One wave handles 16 nodes. Gate pre-activations G = Ht(16x4) @ Wih^T(4x128);
// only gate tiles {i0,i1,g0,g1,o0,o1} are computed (f is dead since c0==0).
// C accumulator is seeded with (b_ih + b_hh) broadcast per gate column.
// A layout: lane m=lane&15, VGPR0/1 = K {2*half, 2*half+1}.
// B layout: lane n=lane&15, VGPR0/1 = K {2*half, 2*half+1} of column (t*16+n).
// D layout: lanes 0-15 hold rows 0..7, lanes 16-31 rows 8..15 (VGPR r -> row).
// ---------------------------------------------------------------------------
__global__ void __launch_bounds__(320)
k_lstm_wmma(const float* __restrict__ ht,
            const float* __restrict__ w_ih,   // [128,4] row-major
            const float* __restrict__ b_ih,   // [128]
            const float* __restrict__ b_hh,   // [128]
            const float* __restrict__ lin_w,  // [32]
            const float* __restrict__ lin_b,  // [1]
            float* __restrict__ out, int n) {
    const int lane = threadIdx.x & 31;
    const int wave = threadIdx.x >> 5;
    const int tile = blockIdx.x * 10 + wave;     // 10 waves / block
    const int base = tile * 16;
    if (base >= n) return;                        // wave-uniform, EXEC stays all-1
    const int col  = lane & 15;
    const int half = lane >> 4;

    // A operand: 16x4 tile of h_tilde
    float4 h4 = ((const float4*)ht)[base + col];
    v2f a;
    a.x = half ? h4.z : h4.x;
    a.y = half ? h4.w : h4.y;

    const int tids[6] = {0, 1, 4, 5, 6, 7};       // i,i, g,g, o,o gate tiles
    v8f acc[6];
#pragma unroll
    for (int q = 0; q < 6; ++q) {
        const int g = tids[q] * 16 + col;         // gate row index 0..127
        v2f b;
        const float* wr = w_ih + g * 4 + 2 * half;
        b.x = wr[0];
        b.y = wr[1];
        const float bs = b_ih[g] + b_hh[g];
        v8f c;
#pragma unroll
        for (int r = 0; r < 8; ++r) c[r] = bs;    // bias baked into accumulator
        acc[q] = __builtin_amdgcn_wmma_f32_16x16x4_f32(
            /*neg_a=*/false, a, /*neg_b=*/false, b,
            /*c_mod=*/(short)0, c, /*reuse_a=*/false, /*reuse_b=*/false);
    }

    // Each lane owns hidden indices {col, 16+col} for its 8 node-rows.
    const float lw1 = lin_w[col];
    const float lw2 = lin_w[16 + col];
    float partial[8];
#pragma unroll
    for (int r = 0; r < 8; ++r) {
        float c1 = fast_sigmoid(acc[0][r]) * fast_tanh(acc[2][r]); // i*tanh(g), j=col
        float c2 = fast_sigmoid(acc[1][r]) * fast_tanh(acc[3][r]); // j=16+col
        float h1 = fast_sigmoid(acc[4][r]) * fast_tanh(c1);
        float h2 = fast_sigmoid(acc[5][r]) * fast_tanh(c2);
        partial[r] = fmaxf(h1, 0.f) * lw1 + fmaxf(h2, 0.f) * lw2;
    }
    // Butterfly-reduce the 16 hidden-index pairs within each 16-lane half.
#pragma unroll
    for (int msk = 1; msk < 16; msk <<= 1) {
#pragma unroll
        for (int r = 0; r < 8; ++r)
            partial[r] += __shfl_xor(partial[r], msk, 32);
    }
    if (col == 0) {
        const float lb = lin_b[0];
        const int node0 = base + half * 8;        // lane0 -> rows 0..7, lane16 -> 8..15
#pragma unroll
        for (int r = 0; r < 8; ++r) out[node0 + r] = partial[r] + lb;
    }
}

// ---------------------------------------------------------------------------
extern "C" void kernel_launch(void* const* d_in, const int* in_sizes, int n_in,
                              void* d_out, int out_size, void* d_ws, size_t ws_size,
                              hipStream_t stream) {
    const float*     x      = (const float*)d_in[0];
    const long long* ei     = (const long long*)d_in[1];   // int64 [2,E]
    const float*     ew     = (const float*)d_in[2];
    const float*     ggc_w  = (const float*)d_in[3];
    const float*     gw_ih  = (const float*)d_in[4];
    const float*     gw_hh  = (const float*)d_in[5];
    const float*     gb_ih  = (const float*)d_in[6];
    const float*     gb_hh  = (const float*)d_in[7];
    const float*     lw_ih  = (const float*)d_in[8];
    // d_in[9] (lstm_w_hh) unused: h0 == 0
    const float*     lb_ih  = (const float*)d_in[10];
    const float*     lb_hh  = (const float*)d_in[11];
    const float*     lin_w  = (const float*)d_in[12];
    const float*     lin_b  = (const float*)d_in[13];
    float*           out    = (float*)d_out;

    const int n = in_sizes[0] / HC;   // 100000
    const int e = in_sizes[2];        // 6400000

    float* ws   = (float*)d_ws;
    float* m    = ws;                 // [N,4]
    float* agg  = ws + 4 * (size_t)n; // [N,4]
    float* cnt  = ws + 8 * (size_t)n; // [N]
    float* htld = ws + 9 * (size_t)n; // [N,4]

    k_ggc_init<<<(n + 255) / 256, 256, 0, stream>>>(x, ggc_w, m, agg, cnt, n);
    k_edge_scatter<<<(e + 255) / 256, 256, 0, stream>>>(ei, ei + e, ew, m, agg, cnt, e);
    k_gru<<<(n + 255) / 256, 256, 0, stream>>>(x, agg, cnt, gw_ih, gw_hh, gb_ih, gb_hh, htld, n);

    const int tiles  = (n + 15) / 16;            // 6250
    const int blocks = (tiles + 9) / 10;         // 625 (10 waves of 32 per block)
    k_lstm_wmma<<<blocks, 320, 0, stream>>>(htld, lw_ih, lb_ih, lb_hh, lin_w, lin_b, out, n);
}